// DiffusioUnpool_60129542144765
// MI455X (gfx1250) — compile-verified
//
#include <hip/hip_runtime.h>

typedef __attribute__((ext_vector_type(2))) float v2f;
typedef __attribute__((ext_vector_type(8))) float v8f;
typedef __attribute__((__vector_size__(16))) int vi4;   // b128 payload type for async builtin

#define ZSTRIDE 144   // 129 cols (128 feat + 1 atte) padded to 9*16
#define NTILES  9
#define Z4      36    // float4 per z row (144/4)

#if defined(__HIP_DEVICE_COMPILE__) && __has_builtin(__builtin_amdgcn_global_load_async_to_lds_b128) && __has_builtin(__builtin_amdgcn_s_wait_asynccnt)
#define USE_ASYNC_LDS 1
#else
#define USE_ASYNC_LDS 0
#endif

// ---------------- init / build kernels ----------------

__global__ void k_init_z(float* __restrict__ z, int total) {
  int i = blockIdx.x * blockDim.x + threadIdx.x;
  if (i < total) z[i] = 0.f;
}

__global__ void k_init_deg_cnt(float* __restrict__ deg, int* __restrict__ cnt, int n) {
  int i = blockIdx.x * blockDim.x + threadIdx.x;
  if (i < n) { deg[i] = 1.f; cnt[i] = 0; }   // 1.0 = self-loop weight
}

__global__ void k_scatter(const float* __restrict__ fea, const float* __restrict__ atte,
                          const int* __restrict__ perm, float* __restrict__ z,
                          int K, int D) {
  int i = blockIdx.x;            // kept node
  int c = threadIdx.x;           // feature
  if (i >= K) return;
  int p = perm[i];
  if (c < D) z[(size_t)p * ZSTRIDE + c] = fea[(size_t)i * D + c];
  if (c == 0) z[(size_t)p * ZSTRIDE + D] = atte[i];
}

__global__ void k_edge_deg(const int* __restrict__ esrc, const float* __restrict__ ew,
                           float* __restrict__ deg, int* __restrict__ cnt, int E) {
  int e = blockIdx.x * blockDim.x + threadIdx.x;
  if (e < E) {
    int s = esrc[e];
    atomicAdd(&deg[s], ew[e]);
    atomicAdd(&cnt[s], 1);
  }
}

__global__ void k_rsqrt(float* __restrict__ deg, int n) {
  int i = blockIdx.x * blockDim.x + threadIdx.x;
  if (i < n) deg[i] = rsqrtf(deg[i]);      // deg -> d^{-1/2} in place
}

// single-block exclusive scan over (cnt[r] + 1) -> row_ptr, cursor
__global__ void k_scan(const int* __restrict__ cnt, int* __restrict__ row_ptr,
                       int* __restrict__ cursor, int n) {
  __shared__ int sdata[1024];
  int t = threadIdx.x;
  int per = (n + 1023) >> 10;
  int r0 = t * per;
  int tot = 0;
  for (int k = 0; k < per; ++k) {
    int r = r0 + k;
    if (r < n) tot += cnt[r] + 1;
  }
  sdata[t] = tot;
  __syncthreads();
  for (int off = 1; off < 1024; off <<= 1) {
    int v = (t >= off) ? sdata[t - off] : 0;
    __syncthreads();
    sdata[t] += v;
    __syncthreads();
  }
  int excl = sdata[t] - tot;
  for (int k = 0; k < per; ++k) {
    int r = r0 + k;
    if (r < n) {
      row_ptr[r] = excl;
      cursor[r]  = excl;
      excl += cnt[r] + 1;
    }
  }
  if (t == 1023) row_ptr[n] = sdata[1023];
}

// fill CSR: real edges + self-loop pseudo-edges (id = E + row)
__global__ void k_fill(const int* __restrict__ esrc, int* __restrict__ cursor,
                       int* __restrict__ edge_perm, int E, int n) {
  int i = blockIdx.x * blockDim.x + threadIdx.x;
  if (i < E) {
    int r = esrc[i];
    int pos = atomicAdd(&cursor[r], 1);
    edge_perm[pos] = i;
  } else if (i < E + n) {
    int r = i - E;
    int pos = atomicAdd(&cursor[r], 1);
    edge_perm[pos] = E + r;
  }
}

// ---------------- WMMA SpMM: one wave per 16-row block, double-buffered ----------------

__global__ void __launch_bounds__(32)
k_spmm_wmma(const float* __restrict__ z, const float* __restrict__ dinv,
            const int* __restrict__ row_ptr, const int* __restrict__ edge_perm,
            const int* __restrict__ esrc, const int* __restrict__ edst,
            const float* __restrict__ ew, float* __restrict__ out,
            int N, int D, int E) {
  __shared__ __align__(16) float lds_z[2][32][ZSTRIDE];  // 2 x 18KB z-row buffers
  __shared__ float lds_cf[2][32];                        // edge coefficients
  __shared__ int   lds_rl[2][32];                        // local row (0..15)
  __shared__ int   lds_dstS[2][32];                      // dst node ids

  const int  lane = threadIdx.x;         // 0..31 (wave32)
  const int  b0   = blockIdx.x * 16;     // first row of this block
  const bool lo   = lane < 16;
  const int  m    = lane & 15;

  int rEnd  = b0 + 16; if (rEnd > N) rEnd = N;
  int start = row_ptr[b0];
  int end   = row_ptr[rEnd];

  v8f acc[NTILES] = {};

  const float4* __restrict__ zg4 = (const float4*)z;

  // stage meta for edges [chunk, chunk+32) into buffer `buf`
  auto stage_meta = [&](int chunk, int buf) {
    int idx = chunk + lane;
    float cf = 0.f; int rl = 0; int dj = 0;
    if (idx < end) {
      int e = edge_perm[idx];
      int s, d; float w;
      if (e >= E) { s = e - E; d = s; w = 1.f; }          // self-loop
      else        { s = esrc[e]; d = edst[e]; w = ew[e]; }
      cf = dinv[s] * w * dinv[d];
      rl = s - b0;
      dj = d;
    }
    lds_cf[buf][lane] = cf; lds_rl[buf][lane] = rl; lds_dstS[buf][lane] = dj;
  };

  // copy the 32 staged z rows (32*36 float4) into lds_z[buf]: 36 all-lane copies
  auto issue_copy = [&](int buf) {
#pragma unroll 4
    for (int i = 0; i < Z4; ++i) {
      int idx = i * 32 + lane;        // 0..1151
      int j   = idx / Z4;             // staged row slot
      int c4  = idx % Z4;             // float4 column
      int row = lds_dstS[buf][j];
      const float4* gp = zg4 + (size_t)row * Z4 + c4;
      float* lp = &lds_z[buf][j][c4 * 4];
#if USE_ASYNC_LDS
      __builtin_amdgcn_global_load_async_to_lds_b128(
          (vi4*)(void*)gp, (vi4*)(void*)lp, 0, 0);
#else
      *(float4*)lp = *gp;
#endif
    }
  };

  if (start < end) {
    stage_meta(start, 0);
    __syncthreads();
    issue_copy(0);

    int buf = 0;
    for (int chunk = start; chunk < end; chunk += 32, buf ^= 1) {
      const int  nxt       = chunk + 32;
      const bool have_next = nxt < end;           // wave-uniform
      if (have_next) {
        stage_meta(nxt, buf ^ 1);
        __syncthreads();
        issue_copy(buf ^ 1);                      // prefetch next chunk
#if USE_ASYNC_LDS
        __builtin_amdgcn_s_wait_asynccnt(Z4);     // <=36 outstanding -> current buf landed
#endif
      } else {
#if USE_ASYNC_LDS
        __builtin_amdgcn_s_wait_asynccnt(0);      // drain for the tail chunk
#endif
      }
      __syncthreads();

      // 8 K-steps of 4 edges; 9 column tiles; f32 WMMA keeps full precision
#pragma unroll
      for (int s4 = 0; s4 < 8; ++s4) {
        const int kb = s4 * 4 + (lo ? 0 : 2);
        v2f a;
        a.x = (lds_rl[buf][kb]     == m) ? lds_cf[buf][kb]     : 0.f;
        a.y = (lds_rl[buf][kb + 1] == m) ? lds_cf[buf][kb + 1] : 0.f;
#pragma unroll
        for (int t = 0; t < NTILES; ++t) {
          const int col = t * 16 + m;
          v2f b;
          b.x = lds_z[buf][kb][col];
          b.y = lds_z[buf][kb + 1][col];
          acc[t] = __builtin_amdgcn_wmma_f32_16x16x4_f32(
              /*neg_a=*/false, a, /*neg_b=*/false, b,
              /*c_mod=*/(short)0, acc[t], /*reuse_a=*/false, /*reuse_b=*/false);
        }
      }
      __syncthreads();
    }
  }

  // epilogue: C layout -> VGPR i: lanes 0-15 hold M=i, lanes 16-31 hold M=8+i
#pragma unroll
  for (int i = 0; i < 8; ++i) {
    int row = b0 + (lo ? i : (8 + i));
    if (row >= N) continue;
#pragma unroll
    for (int t = 0; t < NTILES; ++t) {
      int col = t * 16 + m;
      float v = acc[t][i];
      if (col < D)        out[(size_t)row * D + col] = v;   // features [N, D]
      else if (col == D)  out[(size_t)N * D + row]   = v;   // attention [N, 1]
      // cols D+1..143 are padding -> dropped
    }
  }
}

// ---------------- launcher ----------------

extern "C" void kernel_launch(void* const* d_in, const int* in_sizes, int n_in,
                              void* d_out, int out_size, void* d_ws, size_t ws_size,
                              hipStream_t stream) {
  const float* fea   = (const float*)d_in[0];
  const int*   perm  = (const int*)  d_in[1];
  const int*   eidx  = (const int*)  d_in[2];
  const float* eattr = (const float*)d_in[3];
  const float* atte  = (const float*)d_in[4];
  // d_in[5] = node_num (device scalar, not needed: derive from sizes)

  const int K = in_sizes[1];              // 4096 kept nodes
  const int D = in_sizes[0] / K;          // 128 features
  const int E = in_sizes[3];              // 131072 edges
  const int N = out_size / (D + 1);       // 8192 nodes

  const int* esrc = eidx;
  const int* edst = eidx + E;

  // carve workspace
  char* ws = (char*)d_ws;
  float* z        = (float*)ws; ws += (size_t)N * ZSTRIDE * sizeof(float);
  float* deg      = (float*)ws; ws += (size_t)N * sizeof(float);
  int*   cnt      = (int*)ws;   ws += (size_t)N * sizeof(int);
  int*   row_ptr  = (int*)ws;   ws += (size_t)(N + 1) * sizeof(int);
  int*   cursor   = (int*)ws;   ws += (size_t)N * sizeof(int);
  int*   edge_perm= (int*)ws;   ws += (size_t)(E + N) * sizeof(int);

  const int zTotal = N * ZSTRIDE;
  k_init_z      <<<(zTotal + 255) / 256, 256, 0, stream>>>(z, zTotal);
  k_init_deg_cnt<<<(N + 255) / 256,      256, 0, stream>>>(deg, cnt, N);
  k_scatter     <<<K, D, 0, stream>>>(fea, atte, perm, z, K, D);
  k_edge_deg    <<<(E + 255) / 256,      256, 0, stream>>>(esrc, eattr, deg, cnt, E);
  k_rsqrt       <<<(N + 255) / 256,      256, 0, stream>>>(deg, N);
  k_scan        <<<1, 1024, 0, stream>>>(cnt, row_ptr, cursor, N);
  k_fill        <<<(E + N + 255) / 256,  256, 0, stream>>>(esrc, cursor, edge_perm, E, N);
  k_spmm_wmma   <<<(N + 15) / 16, 32, 0, stream>>>(z, deg, row_ptr, edge_perm,
                                                   esrc, edst, eattr,
                                                   (float*)d_out, N, D, E);
}